// DGCNN_26096221290964
// MI455X (gfx1250) — compile-verified
//
#include <hip/hip_runtime.h>
#include <cstddef>

typedef __attribute__((ext_vector_type(2))) float v2f;
typedef __attribute__((ext_vector_type(8))) float v8f;

#define Bn   32
#define Pn   1024
#define KNN  20
#define BIGF 3.0e38f

// ---------------------------------------------------------------------------
// fp32 WMMA 16x16x4 (codegen-verified last round: v_wmma_f32_16x16x4_f32).
// ---------------------------------------------------------------------------
__device__ __forceinline__ v8f wmma4(v2f a, v2f b, v8f c) {
  return __builtin_amdgcn_wmma_f32_16x16x4_f32(false, a, false, b, (short)0, c,
                                               false, false);
}

// ---------------------------------------------------------------------------
// Build Wcat [kPad, 2H] from W [2C, H]:
//   Wcat[c][h]   = W[c][h] - W[C+c][h]   (A-part: Wt - Wb)
//   Wcat[c][H+h] = W[C+c][h]             (Bv-part: Wb)
// rows c in [C, kPad) zero (K padding for depth-4 WMMA steps).
// ---------------------------------------------------------------------------
__global__ void prep_wcat(const float* __restrict__ W, float* __restrict__ Wcat,
                          int C, int H, int twoH, int kPad) {
  int total = kPad * twoH;
  for (int e = blockIdx.x * blockDim.x + threadIdx.x; e < total;
       e += gridDim.x * blockDim.x) {
    int k = e / twoH, col = e % twoH;
    float v = 0.f;
    if (k < C) {
      int h = (col < H) ? col : (col - H);
      float wt = W[k * H + h];
      float wb = W[(C + k) * H + h];
      v = (col < H) ? (wt - wb) : wb;
    }
    Wcat[e] = v;
  }
}

// XT[g][k][p] = feat[g*Pn+p][k], k in [0,64) zero-padded beyond C.
__global__ void prep_xt(const float* __restrict__ feat, float* __restrict__ XT,
                        int C) {
  int e = blockIdx.x * blockDim.x + threadIdx.x;  // Bn*64*Pn threads exactly
  int p = e & (Pn - 1);
  int k = (e >> 10) & 63;
  int g = e >> 16;
  XT[e] = (k < C) ? feat[(g * Pn + p) * C + k] : 0.f;
}

// sq[node] = ||feat[node]||^2
__global__ void compute_sq(const float* __restrict__ feat,
                           float* __restrict__ sq, int C) {
  int n = blockIdx.x * blockDim.x + threadIdx.x;  // Bn*Pn threads
  const float* r = feat + (size_t)n * C;
  float s = 0.f;
  for (int c = 0; c < C; ++c) s += r[c] * r[c];
  sq[n] = s;
}

// ---------------------------------------------------------------------------
// Fused Gram + kNN top-20. One workgroup (16 waves) per 16-row strip of one
// graph. KSTEPS is compile-time so A-fragments live in registers and are
// reused across the 4 column tiles each wave computes.
// ---------------------------------------------------------------------------
template <int KSTEPS>
__global__ void __launch_bounds__(512) gram_topk(const float* __restrict__ XT,
                                                 const float* __restrict__ sq,
                                                 int* __restrict__ idxOut) {
  __shared__ float Dl[16 * Pn];  // 64 KB

  const int g       = blockIdx.x >> 6;
  const int rowBase = (blockIdx.x & 63) << 4;
  const int wave    = threadIdx.x >> 5;
  const int lane    = threadIdx.x & 31;
  const int l15     = lane & 15;
  const int kkOff   = (lane < 16) ? 0 : 2;
  const int mOff    = (lane < 16) ? 0 : 8;
  const float* Xg   = XT + (size_t)g * 64 * Pn;

  // Hoisted A-fragments: the 16-row strip, all K steps, register-resident.
  v2f af[KSTEPS];
#pragma unroll
  for (int ks = 0; ks < KSTEPS; ++ks) {
    const int kk = (ks << 2) + kkOff;
    af[ks].x = Xg[kk * Pn + rowBase + l15];
    af[ks].y = Xg[(kk + 1) * Pn + rowBase + l15];
  }

  // ---- Phase 1: each wave computes 4 of the 64 16x16 column tiles ----
  for (int ct = wave; ct < 64; ct += 16) {
    const int colBase = ct << 4;
    v8f acc = {0.f, 0.f, 0.f, 0.f, 0.f, 0.f, 0.f, 0.f};
#pragma unroll
    for (int ks = 0; ks < KSTEPS; ++ks) {
      const int kk = (ks << 2) + kkOff;
      v2f b;
      b.x = Xg[kk * Pn + colBase + l15];        // B[kk][n] = X[n][kk]
      b.y = Xg[(kk + 1) * Pn + colBase + l15];
      acc = wmma4(af[ks], b, acc);
    }
    const int colL = colBase + l15;
    const float sn = sq[g * Pn + colL];
#pragma unroll
    for (int v = 0; v < 8; ++v)
      Dl[(v + mOff) * Pn + colL] = sn - 2.f * acc[v];
  }
  __syncthreads();

  // ---- Phase 2: wave w selects KNN smallest for row (rowBase + w) ----
  const int rowL = rowBase + wave;
  float vals[32];
#pragma unroll
  for (int j = 0; j < 32; ++j) {
    int c   = lane + (j << 5);
    float d = Dl[wave * Pn + c];
    vals[j] = (c == rowL) ? BIGF : d;  // exclude self
  }
  const int outBase = (g * Pn + rowL) * KNN;
  for (int sel = 0; sel < KNN; ++sel) {
    float bm = vals[0];
    int bj = 0;
#pragma unroll
    for (int j = 1; j < 32; ++j)
      if (vals[j] < bm) { bm = vals[j]; bj = j; }
    float rv = bm;
    int ri = lane + (bj << 5);
#pragma unroll
    for (int off = 16; off > 0; off >>= 1) {
      float ov = __shfl_xor(rv, off, 32);
      int   oi = __shfl_xor(ri, off, 32);
      if (ov < rv || (ov == rv && oi < ri)) { rv = ov; ri = oi; }
    }
    if (lane == 0) idxOut[outBase + sel] = ri;
#pragma unroll
    for (int j = 0; j < 32; ++j)
      if ((lane + (j << 5)) == ri) vals[j] = BIGF;
  }
}

// ---------------------------------------------------------------------------
// AB[g*Pn+m][0:2H] = X[m] @ Wcat (+bias on first H cols). 4 waves/block; each
// wave sweeps column tiles of one 16-row strip, reusing register A-frags.
// ---------------------------------------------------------------------------
template <int KSTEPS>
__global__ void __launch_bounds__(128) proj_wmma(const float* __restrict__ XT,
                                                 const float* __restrict__ Wcat,
                                                 const float* __restrict__ bias,
                                                 float* __restrict__ AB,
                                                 int twoH, int H) {
  const int nCt  = twoH >> 4;
  const int rt   = blockIdx.x & 63;
  const int g    = blockIdx.x >> 6;
  const int wave = threadIdx.x >> 5;
  const int lane = threadIdx.x & 31;
  const int l15  = lane & 15;
  const int kkOff = (lane < 16) ? 0 : 2;
  const int mOff  = (lane < 16) ? 0 : 8;
  const int rowBase = rt << 4;
  const float* Xg = XT + (size_t)g * 64 * Pn;

  v2f af[KSTEPS];
#pragma unroll
  for (int ks = 0; ks < KSTEPS; ++ks) {
    const int kk = (ks << 2) + kkOff;
    af[ks].x = Xg[kk * Pn + rowBase + l15];
    af[ks].y = Xg[(kk + 1) * Pn + rowBase + l15];
  }

  for (int ct = wave; ct < nCt; ct += 4) {
    const int colBase = ct << 4;
    v8f acc = {0.f, 0.f, 0.f, 0.f, 0.f, 0.f, 0.f, 0.f};
#pragma unroll
    for (int ks = 0; ks < KSTEPS; ++ks) {
      const int kk = (ks << 2) + kkOff;
      v2f b;
      b.x = Wcat[kk * twoH + colBase + l15];
      b.y = Wcat[(kk + 1) * twoH + colBase + l15];
      acc = wmma4(af[ks], b, acc);
    }
    const int col = colBase + l15;
    const float bv = (col < H) ? bias[col] : 0.f;
#pragma unroll
    for (int v = 0; v < 8; ++v)
      AB[(size_t)(g * Pn + rowBase + v + mOff) * twoH + col] = acc[v] + bv;
  }
}

// out[node][h] = relu(A[node][h] + max_j Bv[nbr_j][h])
__global__ void gather_max(const float* __restrict__ AB,
                           const int* __restrict__ idx,
                           float* __restrict__ out, int H, int twoH) {
  int t    = blockIdx.x * blockDim.x + threadIdx.x;  // Bn*Pn*H threads
  int h    = t % H;
  int node = t / H;
  int g    = node >> 10;
  const int* nb = idx + node * KNN;
  float m = -BIGF;
#pragma unroll 4
  for (int j = 0; j < KNN; ++j) {
    int row = (g << 10) + nb[j];
    m = fmaxf(m, AB[(size_t)row * twoH + H + h]);
  }
  float a = AB[(size_t)node * twoH + h];
  out[(size_t)node * H + h] = fmaxf(a + m, 0.f);
}

// pooled[g][h] = max_p h3[g*Pn+p][h]
__global__ void pool_max(const float* __restrict__ h3,
                         float* __restrict__ pooled) {
  int g = blockIdx.x, h = threadIdx.x;  // block = 128
  float m = -BIGF;
  for (int p = 0; p < Pn; ++p)
    m = fmaxf(m, h3[(size_t)((g << 10) + p) * 128 + h]);
  pooled[g * 128 + h] = m;
}

// out[g][h] = relu(bfc[h] + sum_c pooled[g][c] * Wfc[c][h])
__global__ void fc_relu(const float* __restrict__ pooled,
                        const float* __restrict__ Wfc,
                        const float* __restrict__ bfc,
                        float* __restrict__ out) {
  __shared__ float pl[128];
  int g = blockIdx.x, h = threadIdx.x;
  pl[h] = pooled[g * 128 + h];
  __syncthreads();
  float s = bfc[h];
  for (int c = 0; c < 128; ++c) s += pl[c] * Wfc[c * 128 + h];
  out[g * 128 + h] = fmaxf(s, 0.f);
}

// ---------------------------------------------------------------------------
template <int KSTEPS>
static void run_layer(const float* feat, int C, int H, const float* W,
                      const float* bias, float* featOut, float* XT,
                      float* Wcat, float* AB, int* idxB, float* sq,
                      hipStream_t stream) {
  int kPad = KSTEPS * 4;
  int twoH = 2 * H;
  prep_wcat<<<(kPad * twoH + 255) / 256, 256, 0, stream>>>(W, Wcat, C, H, twoH,
                                                           kPad);
  prep_xt<<<(Bn * 64 * Pn) / 256, 256, 0, stream>>>(feat, XT, C);
  compute_sq<<<(Bn * Pn) / 256, 256, 0, stream>>>(feat, sq, C);
  gram_topk<KSTEPS><<<Bn * 64, 512, 0, stream>>>(XT, sq, idxB);
  proj_wmma<KSTEPS><<<Bn * 64, 128, 0, stream>>>(XT, Wcat, bias, AB, twoH, H);
  gather_max<<<(Bn * Pn * H) / 256, 256, 0, stream>>>(AB, idxB, featOut, H,
                                                      twoH);
}

extern "C" void kernel_launch(void* const* d_in, const int* in_sizes, int n_in,
                              void* d_out, int out_size, void* d_ws,
                              size_t ws_size, hipStream_t stream) {
  const float* x   = (const float*)d_in[0];
  const float* W1  = (const float*)d_in[2];
  const float* b1  = (const float*)d_in[3];
  const float* W2  = (const float*)d_in[4];
  const float* b2  = (const float*)d_in[5];
  const float* W3  = (const float*)d_in[6];
  const float* b3  = (const float*)d_in[7];
  const float* Wfc = (const float*)d_in[8];
  const float* bfc = (const float*)d_in[9];
  float* out = (float*)d_out;

  // Workspace layout (~78 MB total, 256B aligned)
  char* ws = (char*)d_ws;
  size_t o = 0;
  auto carve = [&](size_t bytes) {
    void* p = ws + o;
    o = (o + bytes + 255) & ~(size_t)255;
    return p;
  };
  float* h1     = (float*)carve((size_t)Bn * Pn * 64 * 4);
  float* h2     = (float*)carve((size_t)Bn * Pn * 64 * 4);
  float* h3     = (float*)carve((size_t)Bn * Pn * 128 * 4);
  float* XT     = (float*)carve((size_t)Bn * 64 * Pn * 4);
  float* Wcat   = (float*)carve((size_t)64 * 256 * 4);
  float* AB     = (float*)carve((size_t)Bn * Pn * 256 * 4);
  int*   idxB   = (int*)carve((size_t)Bn * Pn * KNN * 4);
  float* sq     = (float*)carve((size_t)Bn * Pn * 4);
  float* pooled = (float*)carve((size_t)Bn * 128 * 4);

  run_layer<1>(x, 3, 64, W1, b1, h1, XT, Wcat, AB, idxB, sq, stream);
  run_layer<16>(h1, 64, 64, W2, b2, h2, XT, Wcat, AB, idxB, sq, stream);
  run_layer<16>(h2, 64, 128, W3, b3, h3, XT, Wcat, AB, idxB, sq, stream);

  pool_max<<<Bn, 128, 0, stream>>>(h3, pooled);
  fc_relu<<<Bn, 128, 0, stream>>>(pooled, Wfc, bfc, out);
}